// BERT_BASE_ATEPC_70746701300093
// MI455X (gfx1250) — compile-verified
//
#include <hip/hip_runtime.h>
#include <hip/hip_bf16.h>

// Problem dims (fixed by the reference)
#define B_  64
#define L_  512
#define D_  768
#define K_  6      // NUM_LABELS
#define P_  3      // POL_DIM

typedef __attribute__((ext_vector_type(2))) float v2f;
typedef __attribute__((ext_vector_type(8))) float v8f;

// ---------------------------------------------------------------------------
// Kernel 0: fill the ate_logits region with b_cls (positions >= nvalid stay
// at bias, matching compacted zeros @ W_cls + b_cls).
// ---------------------------------------------------------------------------
__global__ void init_ate_kernel(float* __restrict__ out, const float* __restrict__ b_cls) {
    int i = blockIdx.x * blockDim.x + threadIdx.x;       // 0 .. B*L*K-1
    if (i < B_ * L_ * K_) {
        out[i] = b_cls[i % K_];
    }
}

// ---------------------------------------------------------------------------
// Kernel 1: per batch row (one wave32): exclusive prefix sum of valid_ids
// (= compaction target index), first-valid index, and gather X0 = first valid
// token (or zeros) into workspace.
// ---------------------------------------------------------------------------
__global__ void scan_gather_kernel(const int* __restrict__ valid,
                                   const float* __restrict__ x,
                                   int* __restrict__ pos,
                                   float* __restrict__ X0) {
    const int b    = blockIdx.x;
    const int lane = threadIdx.x;          // 0..31
    const int base = lane * (L_ / 32);     // 16 tokens per lane

    int v[L_ / 32];
    int sum = 0;
#pragma unroll
    for (int i = 0; i < L_ / 32; ++i) {
        v[i] = valid[b * L_ + base + i];
        sum += v[i];
    }

    // inclusive wave scan of per-lane sums
    int incl = sum;
#pragma unroll
    for (int off = 1; off < 32; off <<= 1) {
        int n = __shfl_up(incl, off, 32);
        if (lane >= off) incl += n;
    }
    int run = incl - sum;                  // exclusive prefix at chunk start

    int firstLocal = 0x7fffffff;
#pragma unroll
    for (int i = 0; i < L_ / 32; ++i) {
        pos[b * L_ + base + i] = run;      // target index if token is valid
        if (v[i] && (base + i) < firstLocal) firstLocal = base + i;
        run += v[i];
    }

    // wave-min to find first valid token index of the row
#pragma unroll
    for (int off = 16; off >= 1; off >>= 1) {
        int o = __shfl_down(firstLocal, off, 32);
        if (o < firstLocal) firstLocal = o;
    }
    firstLocal = __shfl(firstLocal, 0, 32);

    // gather X0[b,:] = x[b, firstValid, :] or zeros
    const float* xrow = x + ((size_t)(b * L_ + firstLocal)) * D_;
    for (int d = lane; d < D_; d += 32) {
        X0[b * D_ + d] = (firstLocal < L_) ? xrow[d] : 0.0f;
    }
}

// ---------------------------------------------------------------------------
// Kernel 2 (bandwidth-critical): stream x once, coalesced float4 loads,
// W_cls in LDS, scatter-write logits to the compacted position.
// Block = 256 threads = 32 tokens x 8 lanes/token.
// ---------------------------------------------------------------------------
__global__ void ate_kernel(const float* __restrict__ x,
                           const int* __restrict__ valid,
                           const int* __restrict__ pos,
                           const float* __restrict__ W_cls,
                           const float* __restrict__ b_cls,
                           float* __restrict__ out) {
    __shared__ float sW[D_ * K_];          // 18 KB
    for (int i = threadIdx.x; i < D_ * K_; i += blockDim.x) sW[i] = W_cls[i];
    __syncthreads();

    const int b     = blockIdx.x >> 4;             // 16 blocks per batch row
    const int tok0  = (blockIdx.x & 15) * 32;
    const int token = tok0 + (threadIdx.x >> 3);
    const int sub   = threadIdx.x & 7;

    const float4* xr = (const float4*)(x + ((size_t)(b * L_ + token)) * D_);

    float acc[K_] = {0.f, 0.f, 0.f, 0.f, 0.f, 0.f};

#pragma unroll 4
    for (int m = 0; m < D_ / 32; ++m) {            // 24 iters, 128B line / 8-lane group
        float4 xv = xr[sub + 8 * m];
        int d = 4 * (sub + 8 * m);
        float xs[4] = {xv.x, xv.y, xv.z, xv.w};
#pragma unroll
        for (int j = 0; j < 4; ++j) {
            const float xj = xs[j];
            const float* w = &sW[(d + j) * K_];
#pragma unroll
            for (int k = 0; k < K_; ++k) acc[k] += xj * w[k];
        }
    }

    // reduce the 8 partial lanes of each token
#pragma unroll
    for (int off = 4; off >= 1; off >>= 1) {
#pragma unroll
        for (int k = 0; k < K_; ++k) acc[k] += __shfl_down(acc[k], off, 8);
    }

    if (sub == 0) {
        const int vi = valid[b * L_ + token];
        if (vi) {
            const int p = pos[b * L_ + token];
            float* o = out + ((size_t)(b * L_ + p)) * K_;
#pragma unroll
            for (int k = 0; k < K_; ++k) o[k] = acc[k] + b_cls[k];
        }
    }
}

// ---------------------------------------------------------------------------
// Kernel 3: BertPooler GEMM via V_WMMA_F32_16X16X4_F32 (full f32 precision).
// pooled = tanh(X0[64,768] @ W_pool[768,768] + b_pool). One wave per 16x16
// tile, grid (64/16, 768/16) = (4, 48). K-loop: 192 steps of 16x16x4.
// ---------------------------------------------------------------------------
__global__ void pooler_wmma_kernel(const float* __restrict__ X0,
                                   const float* __restrict__ W_pool,
                                   const float* __restrict__ b_pool,
                                   float* __restrict__ pooled) {
    const int lane = threadIdx.x;          // 0..31
    const int mn   = lane & 15;            // A row / B col within tile
    const int half = lane >> 4;            // K half-select
    const int row0 = blockIdx.x * 16;
    const int col0 = blockIdx.y * 16;

    const float* arow = X0 + (size_t)(row0 + mn) * D_;

    v8f acc = {};
    for (int k = 0; k < D_; k += 4) {
        // A 16x4 f32: VGPR0 -> K = k + 2*half, VGPR1 -> K = k + 2*half + 1
        v2f a;
        a.x = arow[k + 2 * half];
        a.y = arow[k + 2 * half + 1];
        // B 4x16 f32: same K split; N = lane & 15 (coalesced row reads of W_pool)
        v2f bb;
        bb.x = W_pool[(size_t)(k + 2 * half) * D_ + col0 + mn];
        bb.y = W_pool[(size_t)(k + 2 * half + 1) * D_ + col0 + mn];

        acc = __builtin_amdgcn_wmma_f32_16x16x4_f32(
            /*neg_a=*/false, a, /*neg_b=*/false, bb,
            /*c_mod=*/(short)0, acc, /*reuse_a=*/false, /*reuse_b=*/false);
    }

    // C/D layout: VGPR v -> row = v + 8*half, col = lane & 15
#pragma unroll
    for (int v = 0; v < 8; ++v) {
        const int r = row0 + v + 8 * half;
        const int c = col0 + mn;
        pooled[(size_t)r * D_ + c] = tanhf(acc[v] + b_pool[c]);
    }
}

// ---------------------------------------------------------------------------
// Kernel 4: apc_logits = pooled @ W_apc + b_apc. One wave per batch row.
// ---------------------------------------------------------------------------
__global__ void apc_kernel(const float* __restrict__ pooled,
                           const float* __restrict__ W_apc,
                           const float* __restrict__ b_apc,
                           float* __restrict__ out_apc) {
    const int b    = blockIdx.x;
    const int lane = threadIdx.x;

    float a0 = 0.f, a1 = 0.f, a2 = 0.f;
    for (int d = lane; d < D_; d += 32) {
        const float p = pooled[b * D_ + d];
        const float* w = W_apc + d * P_;
        a0 += p * w[0];
        a1 += p * w[1];
        a2 += p * w[2];
    }
#pragma unroll
    for (int off = 16; off >= 1; off >>= 1) {
        a0 += __shfl_down(a0, off, 32);
        a1 += __shfl_down(a1, off, 32);
        a2 += __shfl_down(a2, off, 32);
    }
    if (lane == 0) {
        float* o = out_apc + b * P_;
        o[0] = a0 + b_apc[0];
        o[1] = a1 + b_apc[1];
        o[2] = a2 + b_apc[2];
    }
}

// ---------------------------------------------------------------------------
extern "C" void kernel_launch(void* const* d_in, const int* in_sizes, int n_in,
                              void* d_out, int out_size, void* d_ws, size_t ws_size,
                              hipStream_t stream) {
    const float* x      = (const float*)d_in[0];   // [B, L, D]
    const int*   valid  = (const int*)  d_in[1];   // [B, L]
    const float* W_cls  = (const float*)d_in[2];   // [D, 6]
    const float* b_cls  = (const float*)d_in[3];   // [6]
    const float* W_pool = (const float*)d_in[4];   // [D, D]
    const float* b_pool = (const float*)d_in[5];   // [D]
    const float* W_apc  = (const float*)d_in[6];   // [D, 3]
    const float* b_apc  = (const float*)d_in[7];   // [3]

    float* out = (float*)d_out;                    // [B*L*6] ++ [B*3]

    // Workspace layout
    int*   pos    = (int*)d_ws;                            // B*L ints   (128 KB)
    float* X0     = (float*)((char*)d_ws + (size_t)B_ * L_ * sizeof(int));
    float* pooled = X0 + (size_t)B_ * D_;                  // B*D floats each

    // 0) bias-fill ate output
    init_ate_kernel<<<(B_ * L_ * K_ + 255) / 256, 256, 0, stream>>>(out, b_cls);

    // 1) per-row scan + first-token gather
    scan_gather_kernel<<<B_, 32, 0, stream>>>(valid, x, pos, X0);

    // 2) ate logits: stream x once, scatter to compacted positions
    ate_kernel<<<B_ * (L_ / 32), 256, 0, stream>>>(x, valid, pos, W_cls, b_cls, out);

    // 3) pooler GEMM on the WMMA pipe (f32)
    pooler_wmma_kernel<<<dim3(B_ / 16, D_ / 16, 1), 32, 0, stream>>>(X0, W_pool, b_pool, pooled);

    // 4) apc head
    apc_kernel<<<B_, 32, 0, stream>>>(pooled, W_apc, b_apc, out + (size_t)B_ * L_ * K_);
}